// EquivariantProductBasisBlock_10033043604095
// MI455X (gfx1250) — compile-verified
//
#include <hip/hip_runtime.h>

// ---------------------------------------------------------------------------
// Compile-time Clebsch-Gordan machinery (replicates the Python reference's
// _su2_cg / _c2r / real_cg exactly, evaluated by clang's constexpr engine).
// All builders are fully self-contained (by-value only, no references to
// other constexpr globals) so they are valid in both host and device passes.
// ---------------------------------------------------------------------------
namespace cgc {

constexpr double csqrt(double x) {
  double g = (x > 1.0) ? x : 1.0;
  for (int i = 0; i < 32; ++i) g = 0.5 * (g + x / g);
  return g;
}
constexpr double dfact(int n) { double r = 1.0; for (int i = 2; i <= n; ++i) r *= (double)i; return r; }
constexpr int imax3(int a, int b, int c) { int m = a; if (b > m) m = b; if (c > m) m = c; return m; }
constexpr int imin3(int a, int b, int c) { int m = a; if (b < m) m = b; if (c < m) m = c; return m; }
constexpr int iabs(int a) { return a < 0 ? -a : a; }
constexpr float fabsf_(float x) { return x < 0.f ? -x : x; }

constexpr double su2_cg(int j1, int m1, int j2, int m2, int j3, int m3) {
  if (m3 != m1 + m2) return 0.0;
  const int vmin = imax3(-j1 + j2 + m3, -j1 + m1, 0);
  const int vmax = imin3(j2 + j3 + m1, j3 - j1 + j2, j3 + m3);
  const double C = csqrt((double)(2 * j3 + 1) * dfact(j3 + j1 - j2) * dfact(j3 - j1 + j2) *
                         dfact(j1 + j2 - j3) * dfact(j3 + m3) * dfact(j3 - m3) /
                         (dfact(j1 + j2 + j3 + 1) * dfact(j1 - m1) * dfact(j1 + m1) *
                          dfact(j2 - m2) * dfact(j2 + m2)));
  double S = 0.0;
  for (int v = vmin; v <= vmax; ++v) {
    const double sgn = ((v + j2 + m2) & 1) ? -1.0 : 1.0;
    S += sgn * dfact(j2 + j3 + m1 - v) * dfact(j1 - m1 + v) /
         (dfact(v) * dfact(j3 - j1 + j2 - v) * dfact(j3 + m3 - v) * dfact(v + j1 - j2 - m3));
  }
  return C * S;
}

struct CD { double re, im; };
constexpr CD cmul(CD a, CD b) { return CD{a.re * b.re - a.im * b.im, a.re * b.im + a.im * b.re}; }
constexpr CD cadd(CD a, CD b) { return CD{a.re + b.re, a.im + b.im}; }
constexpr CD cconj(CD a) { return CD{a.re, -a.im}; }

struct C2R { CD q[7][7]; };
constexpr C2R c2r(int l) {
  C2R r{};
  const double is2 = 1.0 / csqrt(2.0);
  for (int m = -l; m < 0; ++m) {
    r.q[l + m][l - m] = CD{is2, 0.0};     // column l+|m|
    r.q[l + m][l + m] = CD{0.0, -is2};    // column l-|m|
  }
  r.q[l][l] = CD{1.0, 0.0};
  for (int m = 1; m <= l; ++m) {
    const double s = (m & 1) ? -1.0 : 1.0;
    r.q[l + m][l + m] = CD{s * is2, 0.0};
    r.q[l + m][l - m] = CD{0.0, s * is2};
  }
  CD ph{1.0, 0.0};
  for (int t = 0; t < l; ++t) ph = cmul(ph, CD{0.0, -1.0});  // (-i)^l
  for (int a = 0; a < 2 * l + 1; ++a)
    for (int b = 0; b < 2 * l + 1; ++b) r.q[a][b] = cmul(ph, r.q[a][b]);
  return r;
}

struct CGT { float v[7][7][7]; };
constexpr CGT real_cg(int l1, int l2, int l3) {
  CGT out{};
  const C2R q1 = c2r(l1), q2 = c2r(l2), q3 = c2r(l3);
  double Cc[7][7] = {};  // complex CG; n index implied = (i-l1)+(k-l2)+l3
  for (int i = 0; i < 2 * l1 + 1; ++i)
    for (int k = 0; k < 2 * l2 + 1; ++k) {
      const int m1 = i - l1, m2 = k - l2, m3 = m1 + m2;
      if (iabs(m3) <= l3) Cc[i][k] = su2_cg(l1, m1, l2, m2, l3, m3);
    }
  for (int j = 0; j < 2 * l1 + 1; ++j)
    for (int l = 0; l < 2 * l2 + 1; ++l)
      for (int m = 0; m < 2 * l3 + 1; ++m) {
        CD acc{0.0, 0.0};
        for (int i = 0; i < 2 * l1 + 1; ++i)
          for (int k = 0; k < 2 * l2 + 1; ++k) {
            const int n = (i - l1) + (k - l2) + l3;
            if (n < 0 || n > 2 * l3) continue;
            const double cc = Cc[i][k];
            if (cc == 0.0) continue;
            CD t = cmul(q1.q[i][j], q2.q[k][l]);
            t = cmul(t, cconj(q3.q[n][m]));
            acc = cadd(acc, CD{cc * t.re, cc * t.im});
          }
        out.v[j][l][m] = (float)acc.re;
      }
  return out;
}

struct Entry { unsigned char i, j, k, pad; float c; };
struct Meta  { short start; short cnt; unsigned char a1, a2, o, pad; };

constexpr float THRESH = 1e-7f;

// Path counts (match the Python enumeration):
//   PATHS2 = 9, T12_KEYS = 18 (T size 72), PATHS3 = 3*2 + 6*4 + 6*3 + 3*1 = 51
constexpr int NP2 = 9;
constexpr int NTK = 18;
constexpr int NP3 = 51;

// --- Stage 2: correlation-order-2 paths -----------------------------------
struct Stage2 { Entry ent[256]; Meta p[NP2]; int ne; };
constexpr Stage2 build2() {
  Stage2 S{};
  int ne = 0, pi = 0;
  const int sl[3] = {0, 1, 4};
  for (int l1 = 0; l1 < 3; ++l1)
    for (int l2 = 0; l2 < 3; ++l2)
      for (int lo = 0; lo < 2; ++lo) {
        if (!(iabs(l1 - l2) <= lo && lo <= l1 + l2)) continue;
        const CGT g = real_cg(l1, l2, lo);  // by value, self-contained
        const int s = ne;
        for (int a = 0; a < 2*l1+1; ++a)
          for (int b = 0; b < 2*l2+1; ++b)
            for (int c = 0; c < 2*lo+1; ++c)
              if (fabsf_(g.v[a][b][c]) > THRESH)
                S.ent[ne++] = Entry{(unsigned char)a,(unsigned char)b,(unsigned char)c,0, g.v[a][b][c]};
        S.p[pi++] = Meta{(short)s,(short)(ne - s),(unsigned char)sl[l1],(unsigned char)sl[l2],(unsigned char)lo,0};
      }
  S.ne = ne;
  return S;
}
constexpr Stage2 S2 = build2();

// --- Stage T: T12 intermediate couplings (18 keys, 72 values) -------------
struct StageT { Entry ent[1024]; Meta p[NTK]; int ne; };
constexpr StageT buildT() {
  StageT S{};
  int ne = 0, ki = 0, toff = 0;
  const int sl[3] = {0, 1, 4};
  for (int l1 = 0; l1 < 3; ++l1)
    for (int l2 = 0; l2 < 3; ++l2) {
      const int hi = (l1 + l2) < 3 ? (l1 + l2) : 3;
      for (int l12 = iabs(l1 - l2); l12 <= hi; ++l12) {
        const CGT g = real_cg(l1, l2, l12);
        const int s = ne;
        for (int a = 0; a < 2*l1+1; ++a)
          for (int b = 0; b < 2*l2+1; ++b)
            for (int c = 0; c < 2*l12+1; ++c)
              if (fabsf_(g.v[a][b][c]) > THRESH)
                S.ent[ne++] = Entry{(unsigned char)a,(unsigned char)b,(unsigned char)c,0, g.v[a][b][c]};
        S.p[ki] = Meta{(short)s,(short)(ne - s),(unsigned char)sl[l1],(unsigned char)sl[l2],(unsigned char)toff,0};
        toff += 2*l12+1; ++ki;
      }
    }
  S.ne = ne;
  return S;
}
constexpr StageT ST = buildT();

// --- Stage 3: correlation-order-3 paths (51 paths) ------------------------
// Distinct tensors: (0,0,0)(0,1,1)(1,0,1)(1,1,0)(1,1,1)(1,2,1)(2,1,1)(2,2,0)(2,2,1)(3,2,1)
constexpr int slot3(int l12, int l3, int lo) {
  if (l12==0 && l3==0 && lo==0) return 0;
  if (l12==0 && l3==1 && lo==1) return 1;
  if (l12==1 && l3==0 && lo==1) return 2;
  if (l12==1 && l3==1 && lo==0) return 3;
  if (l12==1 && l3==1 && lo==1) return 4;
  if (l12==1 && l3==2 && lo==1) return 5;
  if (l12==2 && l3==1 && lo==1) return 6;
  if (l12==2 && l3==2 && lo==0) return 7;
  if (l12==2 && l3==2 && lo==1) return 8;
  return 9; // (3,2,1)
}
struct Stage3 { Entry ent[1792]; Meta p[NP3]; int ne; };
constexpr Stage3 build3() {
  Stage3 S{};
  // compute the 10 distinct tensors ONCE into locals (by value)
  const CGT gs[10] = {
    real_cg(0,0,0), real_cg(0,1,1), real_cg(1,0,1), real_cg(1,1,0), real_cg(1,1,1),
    real_cg(1,2,1), real_cg(2,1,1), real_cg(2,2,0), real_cg(2,2,1), real_cg(3,2,1)
  };
  // enumerate T12 keys (same order as buildT) to get l12 + T offsets
  int kl12[NTK] = {}; int koff[NTK] = {};
  {
    int ki = 0, toff = 0;
    for (int l1 = 0; l1 < 3; ++l1)
      for (int l2 = 0; l2 < 3; ++l2) {
        const int hi = (l1 + l2) < 3 ? (l1 + l2) : 3;
        for (int l12 = iabs(l1 - l2); l12 <= hi; ++l12) {
          kl12[ki] = l12; koff[ki] = toff; toff += 2*l12+1; ++ki;
        }
      }
  }
  int ne = 0, pi = 0;
  const int sl[3] = {0, 1, 4};
  for (int ki = 0; ki < NTK; ++ki) {
    const int l12 = kl12[ki];
    for (int l3 = 0; l3 < 3; ++l3)
      for (int lo = 0; lo < 2; ++lo) {
        if (!(iabs(l12 - l3) <= lo && lo <= l12 + l3)) continue;
        const int gsi = slot3(l12, l3, lo);
        const int s = ne;
        for (int a = 0; a < 2*l12+1; ++a)
          for (int b = 0; b < 2*l3+1; ++b)
            for (int c = 0; c < 2*lo+1; ++c)
              if (fabsf_(gs[gsi].v[a][b][c]) > THRESH)
                S.ent[ne++] = Entry{(unsigned char)a,(unsigned char)b,(unsigned char)c,0, gs[gsi].v[a][b][c]};
        S.p[pi++] = Meta{(short)s,(short)(ne - s),(unsigned char)koff[ki],(unsigned char)sl[l3],(unsigned char)lo,0};
      }
  }
  S.ne = ne;
  return S;
}
constexpr Stage3 S3 = build3();

} // namespace cgc

// ---------------------------------------------------------------------------
// Constants
// ---------------------------------------------------------------------------
#define N_NODES 4096
#define CHANNELS 128
#define N_FEAT 128
#define PLANE (N_NODES * CHANNELS)   // one B component plane in the workspace

// ---------------------------------------------------------------------------
// Kernel 1: per-(node,channel) CG product basis. Fully unrolled so every
// A[9]/T[72]/B[4] index is a compile-time constant -> pure register FMA chain.
// Writes B planes to workspace as Bws[m][n*128+c] (c contiguous = GEMM K axis).
// ---------------------------------------------------------------------------
__global__ __launch_bounds__(CHANNELS)
void product_basis_kernel(const float* __restrict__ nf, const int* __restrict__ species,
                          const float* __restrict__ w1, const float* __restrict__ w2,
                          const float* __restrict__ w3, float* __restrict__ Bws) {
  const int n = blockIdx.x;
  const int c = threadIdx.x;
  const int e = species[n];

  float A[9];
#pragma unroll
  for (int i = 0; i < 9; ++i) A[i] = nf[(n * CHANNELS + c) * 9 + i];

  float B[4];
  // correlation order 1 (PATHS1 = [0,1])
  B[0] = w1[(e * 2 + 0) * CHANNELS + c] * A[0];
  {
    const float w = w1[(e * 2 + 1) * CHANNELS + c];
    B[1] = w * A[1]; B[2] = w * A[2]; B[3] = w * A[3];
  }
  // correlation order 2
#pragma unroll
  for (int p = 0; p < cgc::NP2; ++p) {
    const float wp = w2[(e * cgc::NP2 + p) * CHANNELS + c];
#pragma unroll
    for (int t = 0; t < cgc::S2.p[p].cnt; ++t) {
      const cgc::Entry E = cgc::S2.ent[cgc::S2.p[p].start + t];
      B[cgc::S2.p[p].o + E.k] += wp * (E.c * A[cgc::S2.p[p].a1 + E.i] * A[cgc::S2.p[p].a2 + E.j]);
    }
  }
  // intermediate couplings T12
  float T[72];
#pragma unroll
  for (int i = 0; i < 72; ++i) T[i] = 0.0f;
#pragma unroll
  for (int q = 0; q < cgc::NTK; ++q) {
#pragma unroll
    for (int t = 0; t < cgc::ST.p[q].cnt; ++t) {
      const cgc::Entry E = cgc::ST.ent[cgc::ST.p[q].start + t];
      T[cgc::ST.p[q].o + E.k] += E.c * A[cgc::ST.p[q].a1 + E.i] * A[cgc::ST.p[q].a2 + E.j];
    }
  }
  // correlation order 3
#pragma unroll
  for (int p = 0; p < cgc::NP3; ++p) {
    const float wp = w3[(e * cgc::NP3 + p) * CHANNELS + c];
#pragma unroll
    for (int t = 0; t < cgc::S3.p[p].cnt; ++t) {
      const cgc::Entry E = cgc::S3.ent[cgc::S3.p[p].start + t];
      B[cgc::S3.p[p].o + E.k] += wp * (E.c * T[cgc::S3.p[p].a1 + E.i] * A[cgc::S3.p[p].a2 + E.j]);
    }
  }

  const int idx = n * CHANNELS + c;
#pragma unroll
  for (int m = 0; m < 4; ++m) Bws[m * PLANE + idx] = B[m];
}

// ---------------------------------------------------------------------------
// Kernel 2: equivariant linear as four fp32 WMMA GEMMs.
// out[n,f,m] = (1/sqrt(128)) * sum_c Bws[m][n,c] * W_m[c,f]
// One wave per 16x16 output tile; V_WMMA_F32_16X16X4_F32, K-loop over C=128.
// Fragment layouts per CDNA5 ISA 7.12.2 (f32 A 16x4, B 4x16, C/D 16x16).
// ---------------------------------------------------------------------------
typedef __attribute__((ext_vector_type(2))) float v2f;
typedef __attribute__((ext_vector_type(8))) float v8f;

__global__ __launch_bounds__(256)
void linear_wmma_kernel(const float* __restrict__ Bws, const float* __restrict__ lw0,
                        const float* __restrict__ lw1, float* __restrict__ out) {
  const int lane = threadIdx.x & 31;
  const int wave = threadIdx.x >> 5;   // f-tile 0..7 (covers F=128)
  const int nt   = blockIdx.x;         // n-tile 0..255
  const int m    = blockIdx.y;         // output irrep component 0..3

  const float* __restrict__ W  = (m == 0) ? lw0 : lw1;
  const float* __restrict__ Bm = Bws + (size_t)m * PLANE;

  const int row = lane & 15;           // A-row / B-col / C-col lane id
  const int hi  = lane >> 4;           // half-wave select (K pair / M+8)
  const int n0  = nt * 16;
  const int f0  = wave * 16;

  // A frag: lanes 0-15 hold K=kb+0,kb+1; lanes 16-31 hold K=kb+2,kb+3 (row = M)
  const float* aPtr = Bm + (size_t)(n0 + row) * CHANNELS + 2 * hi;
  // B frag: VGPR0 = row K=kb+2*hi, VGPR1 = row K=kb+2*hi+1, col = lane&15
  const float* bPtr = W + (size_t)(2 * hi) * N_FEAT + f0 + row;

  v8f acc = {};
  for (int kb = 0; kb < CHANNELS; kb += 4) {
    const v2f a = *(const v2f*)(aPtr + kb);            // 8B aligned
    v2f b;
    b.x = bPtr[(size_t)kb * N_FEAT];
    b.y = bPtr[(size_t)kb * N_FEAT + N_FEAT];
    acc = __builtin_amdgcn_wmma_f32_16x16x4_f32(
        /*neg_a=*/false, a, /*neg_b=*/false, b,
        /*c_mod=*/(short)0, acc, /*reuse_a=*/false, /*reuse_b=*/false);
  }

  const float s = 0.08838834764831845f;  // 1/sqrt(128)
#pragma unroll
  for (int r = 0; r < 8; ++r) {
    const int orow = n0 + r + 8 * hi;    // C/D layout: VGPR r -> M=r (lanes<16) / M=r+8
    const int ocol = f0 + row;
    out[((size_t)orow * N_FEAT + ocol) * 4 + m] = acc[r] * s;
  }
}

// ---------------------------------------------------------------------------
// Entry point
// ---------------------------------------------------------------------------
extern "C" void kernel_launch(void* const* d_in, const int* in_sizes, int n_in,
                              void* d_out, int out_size, void* d_ws, size_t ws_size,
                              hipStream_t stream) {
  (void)in_sizes; (void)n_in; (void)out_size; (void)ws_size;
  const float* nf      = (const float*)d_in[0];  // [4096,128,9]
  const float* w1      = (const float*)d_in[1];  // [10,2,128]
  const float* w2      = (const float*)d_in[2];  // [10,9,128]
  const float* w3      = (const float*)d_in[3];  // [10,51,128]
  const float* lw0     = (const float*)d_in[4];  // [128,128]
  const float* lw1     = (const float*)d_in[5];  // [128,128]
  const int*   species = (const int*)d_in[6];    // [4096]
  float* out = (float*)d_out;                    // [4096,128,4]
  float* Bws = (float*)d_ws;                     // 4 planes of [4096,128] = 8.4 MB

  product_basis_kernel<<<N_NODES, CHANNELS, 0, stream>>>(nf, species, w1, w2, w3, Bws);
  linear_wmma_kernel<<<dim3(N_NODES / 16, 4), 256, 0, stream>>>(Bws, lw0, lw1, out);
}